// GCN_21062519619906
// MI455X (gfx1250) — compile-verified
//
#include <hip/hip_runtime.h>
#include <hip/hip_bf16.h>

typedef __attribute__((ext_vector_type(2))) float v2f;
typedef __attribute__((ext_vector_type(8))) float v8f;

#define IN_CH  512
#define HID_CH 16
#define OUT_CH 40

// ---------------------------------------------------------------- degree prep
__global__ void k_init_deg(float* __restrict__ deg, int n) {
    int i = blockIdx.x * blockDim.x + threadIdx.x;
    if (i < n) deg[i] = 2.0f;                 // two self-loops per node
}

__global__ void k_deg_scatter(const int* __restrict__ ei, float* __restrict__ deg, int E) {
    int e = blockIdx.x * blockDim.x + threadIdx.x;
    if (e < E) atomicAdd(&deg[ei[E + e]], 1.0f);   // dst = ei[E + e]
}

__global__ void k_dinv(float* __restrict__ deg, int n) {
    int i = blockIdx.x * blockDim.x + threadIdx.x;
    if (i < n) {
        float d = deg[i];
        deg[i] = (d > 0.0f) ? rsqrtf(d) : 0.0f;    // in-place deg -> deg^{-1/2}
    }
}

// ------------------------------------------------- GEMM1: xw = x @ W1  (WMMA)
// One wave computes a 16-node x 16-channel tile; K=512 in steps of 4 using
// v_wmma_f32_16x16x4_f32. W1 (512x16 = 32KB) staged in LDS per block.
__global__ void __launch_bounds__(256) k_gemm1(const float* __restrict__ x,
                                               const float* __restrict__ W1,
                                               float* __restrict__ xw,
                                               int nNodes) {
    __shared__ float sW[IN_CH * HID_CH];           // 32 KB
    for (int i = threadIdx.x; i < IN_CH * HID_CH; i += 256) sW[i] = W1[i];
    __syncthreads();

    const int wave = threadIdx.x >> 5;
    const int lane = threadIdx.x & 31;
    const int tile = blockIdx.x * 8 + wave;
    const int nTiles = nNodes >> 4;                // 100000/16 = 6250
    if (tile >= nTiles) return;                    // wave-uniform: EXEC stays all-1s

    const int base = tile << 4;
    const int half = lane >> 4;                    // 0: K-pair {0,1}, 1: {2,3}
    const int l15  = lane & 15;                    // A: row M, B: column N
    const float* __restrict__ xrow = x + (size_t)(base + l15) * IN_CH;

    v8f c = {};
    #pragma unroll 4
    for (int k = 0; k < IN_CH; k += 4) {
        const int ka = k + 2 * half;               // even -> 8B-aligned float2
        v2f a = *(const v2f*)(xrow + ka);          // A[M=l15][ka], A[M=l15][ka+1]
        v2f b;
        b.x = sW[(ka    ) * HID_CH + l15];         // B[ka  ][N=l15]
        b.y = sW[(ka + 1) * HID_CH + l15];         // B[ka+1][N=l15]
        c = __builtin_amdgcn_wmma_f32_16x16x4_f32(
                /*neg_a=*/false, a, /*neg_b=*/false, b,
                /*c_mod=*/(short)0, c, /*reuse_a=*/false, /*reuse_b=*/false);
    }

    // C/D layout: lanes 0-15 -> M = v; lanes 16-31 -> M = v + 8; N = l15
    float* __restrict__ orow = xw + (size_t)base * HID_CH;
    #pragma unroll
    for (int v = 0; v < 8; ++v) {
        int m = v + 8 * half;
        orow[m * HID_CH + l15] = c[v];
    }
}

// ---------------------------------- agg1 = b1 + 2*dinv^2*xw  (self-loop terms)
__global__ void k_init_agg1(const float* __restrict__ xw,
                            const float* __restrict__ dinv,
                            const float* __restrict__ b1,
                            float* __restrict__ agg, int n) {
    int gid = blockIdx.x * blockDim.x + threadIdx.x;
    if (gid < n * HID_CH) {
        int node = gid >> 4, ch = gid & 15;
        float di = dinv[node];
        agg[gid] = b1[ch] + 2.0f * di * di * xw[gid];
    }
}

// ----------------------------- edge scatter, layer 1: 4 threads/edge, float4
__global__ void k_scatter1(const int* __restrict__ ei,
                           const float* __restrict__ dinv,
                           const float* __restrict__ xw,
                           float* __restrict__ agg, int E) {
    int gid = blockIdx.x * blockDim.x + threadIdx.x;
    if (gid >= E * 4) return;
    int e  = gid >> 2;
    int c0 = (gid & 3) << 2;
    int s = ei[e], d = ei[E + e];
    float norm = dinv[s] * dinv[d];
    const float4 v = *(const float4*)(xw + (size_t)s * HID_CH + c0);
    float* ap = agg + (size_t)d * HID_CH + c0;
    atomicAdd(ap + 0, v.x * norm);
    atomicAdd(ap + 1, v.y * norm);
    atomicAdd(ap + 2, v.z * norm);
    atomicAdd(ap + 3, v.w * norm);
}

// ------------------------- GEMM2: xw2 = relu(agg1) @ W2  (tiny: 128 MFLOP)
__global__ void __launch_bounds__(256) k_gemm2(const float* __restrict__ agg,
                                               const float* __restrict__ W2,
                                               float* __restrict__ xw2, int n) {
    __shared__ float sW[HID_CH * OUT_CH];          // 2.5 KB
    for (int i = threadIdx.x; i < HID_CH * OUT_CH; i += 256) sW[i] = W2[i];
    __syncthreads();
    int gid = blockIdx.x * 256 + threadIdx.x;
    if (gid >= n * OUT_CH) return;
    int node = gid / OUT_CH;
    int oc   = gid - node * OUT_CH;
    const float* __restrict__ arow = agg + (size_t)node * HID_CH;
    float sum = 0.0f;
    #pragma unroll
    for (int k = 0; k < HID_CH; ++k)
        sum = fmaf(fmaxf(arow[k], 0.0f), sW[k * OUT_CH + oc], sum);
    xw2[gid] = sum;
}

// -------------------------------- d_out = b2 + 2*dinv^2*xw2 (self-loop terms)
__global__ void k_init_out(const float* __restrict__ xw2,
                           const float* __restrict__ dinv,
                           const float* __restrict__ b2,
                           float* __restrict__ out, int n) {
    int gid = blockIdx.x * blockDim.x + threadIdx.x;
    if (gid < n * OUT_CH) {
        int node = gid / OUT_CH;
        int ch   = gid - node * OUT_CH;
        float di = dinv[node];
        out[gid] = b2[ch] + 2.0f * di * di * xw2[gid];
    }
}

// ---------------------------- edge scatter, layer 2: 10 threads/edge, float4
__global__ void k_scatter2(const int* __restrict__ ei,
                           const float* __restrict__ dinv,
                           const float* __restrict__ xw2,
                           float* __restrict__ out, int E) {
    int gid = blockIdx.x * blockDim.x + threadIdx.x;
    if (gid >= E * 10) return;
    int e  = gid / 10;
    int c0 = (gid - e * 10) << 2;                  // 0,4,...,36
    int s = ei[e], d = ei[E + e];
    float norm = dinv[s] * dinv[d];
    const float4 v = *(const float4*)(xw2 + (size_t)s * OUT_CH + c0);
    float* ap = out + (size_t)d * OUT_CH + c0;
    atomicAdd(ap + 0, v.x * norm);
    atomicAdd(ap + 1, v.y * norm);
    atomicAdd(ap + 2, v.z * norm);
    atomicAdd(ap + 3, v.w * norm);
}

// ---------------------------------------------------------------------- launch
extern "C" void kernel_launch(void* const* d_in, const int* in_sizes, int n_in,
                              void* d_out, int out_size, void* d_ws, size_t ws_size,
                              hipStream_t stream) {
    const float* x   = (const float*)d_in[0];
    const int*   ei  = (const int*)d_in[1];
    const float* W1  = (const float*)d_in[2];
    const float* b1  = (const float*)d_in[3];
    const float* W2  = (const float*)d_in[4];
    const float* b2  = (const float*)d_in[5];
    float* out = (float*)d_out;

    const int N = in_sizes[0] / IN_CH;     // 100000
    const int E = in_sizes[1] / 2;         // 3200000

    // Workspace layout (all offsets 16B-aligned given N % 4 == 0)
    char* ws   = (char*)d_ws;
    float* dinv = (float*)ws;                                       // N
    float* xw1  = (float*)(ws + (size_t)N * 4);                     // N*16
    float* agg1 = (float*)(ws + (size_t)N * 4 + (size_t)N * 64);    // N*16
    float* xw2  = (float*)(ws + (size_t)N * 4 + (size_t)N * 128);   // N*40

    const int B = 256;
    auto cdiv = [](long long a, long long b) { return (int)((a + b - 1) / b); };

    // 1) degree -> dinv
    k_init_deg   <<<cdiv(N, B), B, 0, stream>>>(dinv, N);
    k_deg_scatter<<<cdiv(E, B), B, 0, stream>>>(ei, dinv, E);
    k_dinv       <<<cdiv(N, B), B, 0, stream>>>(dinv, N);

    // 2) layer 1: WMMA projection, self-loop/bias init, edge scatter
    k_gemm1    <<<cdiv((N >> 4), 8), B, 0, stream>>>(x, W1, xw1, N);
    k_init_agg1<<<cdiv((long long)N * HID_CH, B), B, 0, stream>>>(xw1, dinv, b1, agg1, N);
    k_scatter1 <<<cdiv((long long)E * 4, B), B, 0, stream>>>(ei, dinv, xw1, agg1, E);

    // 3) layer 2: relu fused into tiny GEMM, self-loop/bias init, edge scatter
    k_gemm2   <<<cdiv((long long)N * OUT_CH, B), B, 0, stream>>>(agg1, W2, xw2, N);
    k_init_out<<<cdiv((long long)N * OUT_CH, B), B, 0, stream>>>(xw2, dinv, b2, out, N);
    k_scatter2<<<cdiv((long long)E * 10, B), B, 0, stream>>>(ei, dinv, xw2, out, E);
}